// GatedGCNLayer_27900107555156
// MI455X (gfx1250) — compile-verified
//
#include <hip/hip_runtime.h>

typedef __attribute__((ext_vector_type(2))) float v2f;
typedef __attribute__((ext_vector_type(8))) float v8f;

#define D 128
#define TILE_M 16

// Fused 4-way node-projection GEMM. blockIdx.y selects which projection:
//   0: A = h @ Wg0            -> ws[0]
//   1: B = h @ Wg1 + bg       -> ws[1]
//   2: C = h @ Ws  + bs       -> ws[2]
//   3: R = h @ Wr  + br       -> d_out (edge kernel atomically accumulates on top)
__global__ __launch_bounds__(256) void gemm4_kernel(
    const float* __restrict__ h,
    const float* __restrict__ Wg,
    const float* __restrict__ Ws,
    const float* __restrict__ Wr,
    const float* __restrict__ bg,
    const float* __restrict__ bs,
    const float* __restrict__ br,
    float* __restrict__ Aout, float* __restrict__ Bout,
    float* __restrict__ Cout, float* __restrict__ Rout,
    int N)
{
  __shared__ float sh[TILE_M * 129];        // padded stride: bank = (m + k) % 64
  const int m0  = blockIdx.x * TILE_M;
  const int wid = blockIdx.y;

  const float* W; const float* bias; float* dst;
  switch (wid) {
    case 0:  W = Wg;         bias = nullptr; dst = Aout; break;
    case 1:  W = Wg + D * D; bias = bg;      dst = Bout; break;
    case 2:  W = Ws;         bias = bs;      dst = Cout; break;
    default: W = Wr;         bias = br;      dst = Rout; break;
  }

  // Stage 16x128 tile of h into LDS (coalesced global reads).
  for (int i = threadIdx.x; i < TILE_M * D; i += 256) {
    int r = i >> 7, c = i & (D - 1);
    int row = m0 + r; if (row >= N) row = N - 1;
    sh[r * 129 + c] = h[(long long)row * D + c];
  }
  __syncthreads();

  const int lane = threadIdx.x & 31;
  const int wv   = threadIdx.x >> 5;        // wave id 0..7 -> 16-col output slice
  const int mrow = lane & 15;               // A: M index / B: N index
  const int kh   = (lane >> 4) << 1;        // K sub-offset: 0 or 2
  const int col  = wv * 16 + mrow;

  v8f acc = {0.f, 0.f, 0.f, 0.f, 0.f, 0.f, 0.f, 0.f};
  #pragma unroll 4
  for (int k0 = 0; k0 < D; k0 += 4) {
    v2f a, b;
    a.x = sh[mrow * 129 + k0 + kh];                 // A[m][k], k = k0+kh(+1)
    a.y = sh[mrow * 129 + k0 + kh + 1];
    b.x = W[(k0 + kh) * D + col];                   // B[k][n] (W row-major DxD)
    b.y = W[(k0 + kh + 1) * D + col];
    acc = __builtin_amdgcn_wmma_f32_16x16x4_f32(
        /*neg_a=*/false, a, /*neg_b=*/false, b,
        /*c_mod=*/(short)0, acc, /*reuse_a=*/false, /*reuse_b=*/false);
  }

  const float bv    = bias ? bias[col] : 0.f;       // per output-feature bias
  const int   mbase = m0 + ((lane >> 4) << 3);      // C/D layout: M = vgpr + 8*(lane/16)

  // Per-lane base address; consecutive accumulator rows are +D floats (=512B),
  // so the fast path lowers to 8 global_store_b32 with immediate offsets.
  float* p = dst + (long long)mbase * D + col;
  if (m0 + TILE_M <= N) {                           // uniform: whole tile in range
    #pragma unroll
    for (int i = 0; i < 8; ++i) p[(size_t)i * D] = acc[i] + bv;
  } else {                                          // ragged tail (never hit when N%16==0)
    #pragma unroll
    for (int i = 0; i < 8; ++i)
      if (mbase + i < N) p[(size_t)i * D] = acc[i] + bv;
  }
}

// One wave per edge; each lane owns 4 features (32 lanes x float4 = 128).
// All gathers are 512B contiguous rows; A/B/C (153 MB total) stay L2-resident.
__global__ __launch_bounds__(256) void edge_kernel(
    const float* __restrict__ Ag,        // h@Wg0        [N,D]
    const float* __restrict__ Bg,        // h@Wg1 + bg   [N,D]
    const float* __restrict__ Cg,        // h@Ws  + bs   [N,D]
    const long long* __restrict__ ei,    // [2,E] int64: row0=send, row1=rec
    float* __restrict__ out,             // pre-filled with h@Wr + br
    int E)
{
  int gid  = blockIdx.x * blockDim.x + threadIdx.x;
  int e    = gid >> 5;
  int lane = gid & 31;
  if (e >= E) return;

  long long send = ei[e];
  long long rec  = ei[(long long)E + e];

  const float4 a = ((const float4*)(Ag + rec  * (long long)D))[lane];
  const float4 b = ((const float4*)(Bg + send * (long long)D))[lane];
  const float4 c = ((const float4*)(Cg + send * (long long)D))[lane];

  float4 m;
  m.x = c.x * (1.f / (1.f + __expf(-(a.x + b.x))));
  m.y = c.y * (1.f / (1.f + __expf(-(a.y + b.y))));
  m.z = c.z * (1.f / (1.f + __expf(-(a.z + b.z))));
  m.w = c.w * (1.f / (1.f + __expf(-(a.w + b.w))));

  float* o = out + rec * (long long)D + lane * 4;
  atomicAdd(o + 0, m.x);
  atomicAdd(o + 1, m.y);
  atomicAdd(o + 2, m.z);
  atomicAdd(o + 3, m.w);
}

extern "C" void kernel_launch(void* const* d_in, const int* in_sizes, int n_in,
                              void* d_out, int out_size, void* d_ws, size_t ws_size,
                              hipStream_t stream) {
  const float*     h  = (const float*)d_in[0];
  const long long* ei = (const long long*)d_in[1];   // int64 edge_index [2,E]
  const float*     Wg = (const float*)d_in[2];       // [2D, D]
  const float*     bg = (const float*)d_in[3];
  const float*     Ws = (const float*)d_in[4];       // [D, D]
  const float*     bs = (const float*)d_in[5];
  const float*     Wr = (const float*)d_in[6];       // [D, D]
  const float*     br = (const float*)d_in[7];

  const int N = in_sizes[0] / D;      // 100000
  const int E = in_sizes[1] / 2;      // 600000

  float* Aout = (float*)d_ws;                        // 3 * N * D * 4B = 153.6 MB
  float* Bout = Aout + (size_t)N * D;
  float* Cout = Bout + (size_t)N * D;
  float* out  = (float*)d_out;

  dim3 g1((N + TILE_M - 1) / TILE_M, 4);
  gemm4_kernel<<<g1, 256, 0, stream>>>(h, Wg, Ws, Wr, bg, bs, br,
                                       Aout, Bout, Cout, out, N);

  int blocks = (E * 32 + 255) / 256;   // 8 edges (waves) per 256-thread block
  edge_kernel<<<blocks, 256, 0, stream>>>(Aout, Bout, Cout, ei, out, E);
}